// MicrofacetBase_6073083756908
// MI455X (gfx1250) — compile-verified
//
#include <hip/hip_runtime.h>
#include <cstdint>

#define BLOCK 256
#define TILE  256   // elements per block; 256*36B = 9216B input tile

typedef float v4f __attribute__((ext_vector_type(4)));

__global__ __launch_bounds__(BLOCK) void MicrofacetBase_kernel(
    const float* __restrict__ in,          // [n,3,3]
    const float* __restrict__ base_color,  // [3]
    const float* __restrict__ eta_p,       // [1]
    float* __restrict__ out,               // [n,3]
    int n)
{
    __shared__ __align__(16) float s_in[TILE * 9];   // 9216 B
    __shared__ __align__(16) float s_out[TILE * 3];  // 3072 B

    const int tid = threadIdx.x;
    const long long tile_base = (long long)blockIdx.x * TILE;
    const bool full = (tile_base + TILE) <= (long long)n;
    const long long gi = tile_base + tid;

    const float e   = eta_p[0];
    const float bc0 = base_color[0];
    const float bc1 = base_color[1];
    const float bc2 = base_color[2];

    float l0, l1, l2, nn0, nn1, nn2, v0, v1, v2;

    if (full) {
        // CDNA5 async global->LDS staging of the 9216B tile as 576 b128
        // copies in exactly 3 instructions: INST_OFFSET is added to BOTH the
        // LDS dest and the global source, and our deltas are identical
        // (256 lanes * 16B = 4096B per step). Data is single-use -> NT hint.
        const uint64_t gaddr =
            (uint64_t)(const char*)in + (uint64_t)tile_base * 36ull + (uint64_t)tid * 16ull;
        const uint32_t lds_addr = (uint32_t)(uintptr_t)(&s_in[tid * 4]);

        asm volatile("global_load_async_to_lds_b128 %0, %1, off th:TH_LOAD_NT"
                     :: "v"(lds_addr), "v"(gaddr) : "memory");
        asm volatile("global_load_async_to_lds_b128 %0, %1, off offset:4096 th:TH_LOAD_NT"
                     :: "v"(lds_addr), "v"(gaddr) : "memory");
        if (tid < 64) {
            asm volatile("global_load_async_to_lds_b128 %0, %1, off offset:8192 th:TH_LOAD_NT"
                         :: "v"(lds_addr), "v"(gaddr) : "memory");
        }
        asm volatile("s_wait_asynccnt 0" ::: "memory");
        __syncthreads();

        const float* p = &s_in[tid * 9];  // stride-9 dwords: conflict-free on 64 banks
        l0  = p[0]; l1  = p[1]; l2  = p[2];
        nn0 = p[3]; nn1 = p[4]; nn2 = p[5];
        v0  = p[6]; v1  = p[7]; v2  = p[8];
    } else {
        // Tail block (not exercised for N=4M, kept for generality)
        if (gi < n) {
            const float* p = in + gi * 9;
            l0  = p[0]; l1  = p[1]; l2  = p[2];
            nn0 = p[3]; nn1 = p[4]; nn2 = p[5];
            v0  = p[6]; v1  = p[7]; v2  = p[8];
        } else {
            l0=l1=l2=nn0=nn1=nn2=v0=v1=v2 = 0.0f;
        }
    }

    // ---- microfacet math (faithful to reference; d_term is literally zero) ----
    float hv0 = l0 + v0, hv1 = l1 + v1, hv2 = l2 + v2;
    float hnorm = sqrtf(hv0 * hv0 + hv1 * hv1 + hv2 * hv2);
    hnorm = fmaxf(hnorm, 1e-12f);

    float nl = nn0 * l0 + nn1 * l1 + nn2 * l2;
    float nv = nn0 * v0 + nn1 * v1 + nn2 * v2;
    // half only ever appears inside dot(view, half): fold the normalization in.
    float vh = (v0 * hv0 + v1 * hv1 + v2 * hv2) / hnorm;

    float gterm = nl * nv;
    float c  = vh;
    float gg = e * e + c * c - 1.0f;
    float gs = sqrtf(fmaxf(gg, 1e-12f));
    float A  = (gs - c) / (gs + c);
    float B  = (c * (gs + c) - 1.0f) / (c * (gs - c) + 1.0f);
    float fr = (gg > 0.0f) ? (0.5f * A * A * (1.0f + B * B)) : 1.0f;
    float denom  = 4.0f * nl * nv;
    float d_term = 0.0f;                        // zeros_like(vh) in the reference
    float scale  = d_term * gterm * fr / denom; // preserves NaN/sign semantics

    if (full) {
        // Stage output through LDS -> one dense NT b128 store per lane.
        s_out[tid * 3 + 0] = bc0 * scale;
        s_out[tid * 3 + 1] = bc1 * scale;
        s_out[tid * 3 + 2] = bc2 * scale;
        __syncthreads();

        if (tid < TILE * 3 / 4) {  // 192 lanes cover 3072B
            v4f val = *(const v4f*)&s_out[tid * 4];
            v4f* gdst = (v4f*)((char*)out + (uint64_t)tile_base * 12ull
                                          + (uint64_t)tid * 16ull);
            __builtin_nontemporal_store(val, gdst);
        }
    } else if (gi < n) {
        out[gi * 3 + 0] = bc0 * scale;
        out[gi * 3 + 1] = bc1 * scale;
        out[gi * 3 + 2] = bc2 * scale;
    }
}

extern "C" void kernel_launch(void* const* d_in, const int* in_sizes, int n_in,
                              void* d_out, int out_size, void* d_ws, size_t ws_size,
                              hipStream_t stream) {
    const float* in         = (const float*)d_in[0];
    const float* base_color = (const float*)d_in[1];
    // d_in[2] = alpha (unused by the reference math)
    const float* eta        = (const float*)d_in[3];

    const int n    = in_sizes[0] / 9;
    const int grid = (n + TILE - 1) / TILE;

    MicrofacetBase_kernel<<<grid, BLOCK, 0, stream>>>(
        in, base_color, eta, (float*)d_out, n);
}